// AdaptiveLogSoftmax_84499186581598
// MI455X (gfx1250) — compile-verified
//
#include <hip/hip_runtime.h>
#include <math.h>

// ---------------------------------------------------------------------------
// Adaptive log-softmax NLL for MI455X (gfx1250, wave32, WMMA + async-LDS).
//
//   hidden [1024,1024] f32, target [1024] i32
//   cluster 0 (head): W0 [20003,1024]  b0 p0[1024,1024]
//   cluster 1:        W1 [20000, 256]  b1 p1[1024, 256]
//   cluster 2:        W2 [160000, 64]  b2 p2[1024,  64]
//   cluster 3:        W3 [67735,  16]  b3 p3[1024,  16]
//
// Pipeline:
//   A) proj_c = hidden @ p_c            (f32 WMMA 16x16x4 GEMM -> ws)
//   B) per cluster: W tiles staged once per block into LDS with
//      GLOBAL_LOAD_ASYNC_TO_LDS_B128 (double buffered, s_wait_asynccnt),
//      logitsᵀ tiles D = W_tile(16xK) x projᵀ(Kx16) with two token tiles
//      per wave (A-fragment reuse), fused online softmax, target gather,
//      head cluster-column gather; partials per (chunk, token) -> ws
//   C) reduce partials -> per-token NLL
// ---------------------------------------------------------------------------

typedef float v2f __attribute__((ext_vector_type(2)));
typedef float v8f __attribute__((ext_vector_type(8)));

#define NTOK   1024
#define DPROJ  1024
#define NCHUNK 32            // vocab chunks per cluster (grid.x)

__device__ __forceinline__ v8f wmma_f32x4(v2f a, v2f b, v8f c) {
  return __builtin_amdgcn_wmma_f32_16x16x4_f32(
      /*neg_a=*/false, a, /*neg_b=*/false, b,
      /*c_mod=*/(short)0, c, /*reuse_a=*/false, /*reuse_b=*/false);
}

__device__ __forceinline__ void online_upd(float& m, float& s, float x) {
  if (x > m) { s = s * __expf(m - x) + 1.f; m = x; }
  else       { s += __expf(x - m); }
}

__device__ __forceinline__ void online_merge(float& m, float& s,
                                             float m2, float s2) {
  const float nm = fmaxf(m, m2);
  s = (nm == -INFINITY) ? 0.f
                        : (s * __expf(m - nm) + s2 * __expf(m2 - nm));
  m = nm;
}

// async copy of 16 bytes global -> LDS (ASYNCcnt-tracked, wave-level)
__device__ __forceinline__ void async_g2l_b128(const float* g, unsigned lds) {
  asm volatile("global_load_async_to_lds_b128 %0, %1, off"
               :: "v"(lds), "v"(g) : "memory");
}

// ---------------------------------------------------------------------------
// Phase A: proj[t][j] = sum_k hidden[t][k] * P[k][j]; one wave per 16x16 tile.
// ---------------------------------------------------------------------------
__global__ void proj_gemm(const float* __restrict__ H,
                          const float* __restrict__ P,
                          float* __restrict__ out, int d) {
  const int lane = threadIdx.x & 31;
  const int lr = lane & 15;
  const int hi = lane >> 4;
  const int t0 = blockIdx.x * 16;
  const int j0 = blockIdx.y * 16;

  const float* hrow = H + (size_t)(t0 + lr) * DPROJ;
  v8f acc = {};
#pragma unroll 4
  for (int k = 0; k < DPROJ; k += 4) {
    v2f a = *(const v2f*)(hrow + k + 2 * hi);
    v2f b;
    b.x = P[(size_t)(k + 2 * hi) * d + j0 + lr];
    b.y = P[(size_t)(k + 2 * hi + 1) * d + j0 + lr];
    acc = wmma_f32x4(a, b, acc);
  }
#pragma unroll
  for (int r = 0; r < 8; ++r)
    out[(size_t)(t0 + r + 8 * hi) * d + j0 + lr] = acc[r];
}

// ---------------------------------------------------------------------------
// Phase B: per-cluster streamed logits + online softmax.
// Block = 128 threads (4 waves); wave w owns tokens [by*128 + 32w, +32)
// as TWO 16-token tiles sharing every A (W) fragment.
// For D >= 64 the W tile is staged once per block into LDS via async copy
// (double buffered); A-fragments then come from conflict-free ds_load_b64.
// ---------------------------------------------------------------------------
template <int D>
__global__ void cluster_pass(const float* __restrict__ W,
                             const float* __restrict__ bias,
                             const float* __restrict__ proj,
                             const int*  __restrict__ target,
                             float* __restrict__ part,
                             int size, int base,
                             int tiles, int chunkTiles,
                             int isHead, int chunkBase) {
  constexpr bool USE_LDS = (D >= 64);
  constexpr int  ROWF    = D + 4;                 // padded row (floats)
  constexpr int  TRANS   = 4 * D;                 // b128 copies per tile
  constexpr int  LPS     = (TRANS + 127) / 128;   // async ops per wave/stage

  __shared__ float smem[USE_LDS ? 2 * 16 * ROWF : 1];

  const int tid  = threadIdx.x;
  const int lane = tid & 31;
  const int wv   = tid >> 5;
  const int lr   = lane & 15;
  const int hi   = lane >> 4;

  const int tokenA = blockIdx.y * 128 + wv * 32 + lr;
  const int tokenB = tokenA + 16;

  const float* browA = proj + (size_t)tokenA * D;
  const float* browB = proj + (size_t)tokenB * D;

  const int tA = target[tokenA];
  const int tB = target[tokenB];
  const int trelA = isHead ? min(tA, size - 4)
                           : min(max(tA - base, 0), size - 1);
  const int trelB = isHead ? min(tB, size - 4)
                           : min(max(tB - base, 0), size - 1);

  float mA = -INFINITY, sA = 0.f, tlA = -INFINITY;
  float mB = -INFINITY, sB = 0.f, tlB = -INFINITY;
  float rcA[3] = {-INFINITY, -INFINITY, -INFINITY};
  float rcB[3] = {-INFINITY, -INFINITY, -INFINITY};

  const int vtBeg = blockIdx.x * chunkTiles;
  const int vtEnd = min(vtBeg + chunkTiles, tiles);

  // stage W tile vt into LDS buffer buf (all 128 threads cooperate)
  auto stage = [&](int buf, int vt) {
    const int v0 = vt << 4;
    for (int idx = tid; idx < TRANS; idx += 128) {
      const int row = idx / (D / 4);
      const int kq  = idx - row * (D / 4);
      const int vr  = min(v0 + row, size - 1);
      const float*   g   = W + (size_t)vr * D + kq * 4;
      const unsigned lds = (unsigned)(uintptr_t)&smem[buf * 16 * ROWF +
                                                      row * ROWF + kq * 4];
      async_g2l_b128(g, lds);
    }
  };

  if (USE_LDS && vtBeg < vtEnd) stage(0, vtBeg);

  for (int vt = vtBeg; vt < vtEnd; ++vt) {
    const int v0  = vt << 4;
    const int cur = (vt - vtBeg) & 1;
    const float* wrow = nullptr;
    const float* lrow = nullptr;

    if (USE_LDS) {
      const bool more = (vt + 1) < vtEnd;
      if (more) {
        stage(cur ^ 1, vt + 1);
        asm volatile("s_wait_asynccnt %0" :: "n"(LPS) : "memory");
      } else {
        asm volatile("s_wait_asynccnt 0" ::: "memory");
      }
      __syncthreads();                         // stage(cur) visible to all
      lrow = &smem[cur * 16 * ROWF + lr * ROWF];
    } else {
      const int vr = min(v0 + lr, size - 1);
      wrow = W + (size_t)vr * D;
    }

    v8f acc0 = {}, acc1 = {};
#pragma unroll 4
    for (int k = 0; k < D; k += 4) {
      v2f a;
      if (USE_LDS) a = *(const v2f*)(lrow + k + 2 * hi);   // ds_load_b64
      else         a = *(const v2f*)(wrow + k + 2 * hi);
      v2f b0 = *(const v2f*)(browA + k + 2 * hi);
      v2f b1 = *(const v2f*)(browB + k + 2 * hi);
      acc0 = wmma_f32x4(a, b0, acc0);                      // A reused x2
      acc1 = wmma_f32x4(a, b1, acc1);
    }

    if (USE_LDS) __syncthreads();              // done reading before overwrite

#pragma unroll
    for (int r = 0; r < 8; ++r) {
      const int v = v0 + hi * 8 + r;
      if (v < size) {
        const float bv = bias[v];
        const float xA = acc0[r] + bv;
        const float xB = acc1[r] + bv;
        online_upd(mA, sA, xA);
        online_upd(mB, sB, xB);
        if (v == trelA) tlA = xA;
        if (v == trelB) tlB = xB;
        if (isHead && v >= size - 3) {         // head cols 20000..20002
          const int ci = v - (size - 3);
          rcA[ci] = xA;
          rcB[ci] = xB;
        }
      }
    }
  }

  // Combine the two lanes (lane, lane^16) covering each token's vocab halves.
  online_merge(mA, sA, __shfl_xor(mA, 16), __shfl_xor(sA, 16));
  online_merge(mB, sB, __shfl_xor(mB, 16), __shfl_xor(sB, 16));
  tlA = fmaxf(tlA, __shfl_xor(tlA, 16));
  tlB = fmaxf(tlB, __shfl_xor(tlB, 16));
#pragma unroll
  for (int c = 0; c < 3; ++c) {
    rcA[c] = fmaxf(rcA[c], __shfl_xor(rcA[c], 16));
    rcB[c] = fmaxf(rcB[c], __shfl_xor(rcB[c], 16));
  }

  if (lane < 16) {
    float* pA = part + ((size_t)(chunkBase + blockIdx.x) * NTOK + tokenA) * 6;
    pA[0] = mA; pA[1] = sA; pA[2] = tlA;
    pA[3] = rcA[0]; pA[4] = rcA[1]; pA[5] = rcA[2];
    float* pB = part + ((size_t)(chunkBase + blockIdx.x) * NTOK + tokenB) * 6;
    pB[0] = mB; pB[1] = sB; pB[2] = tlB;
    pB[3] = rcB[0]; pB[4] = rcB[1]; pB[5] = rcB[2];
  }
}

// ---------------------------------------------------------------------------
// Phase C: reduce chunk partials -> per-token NLL.
// ---------------------------------------------------------------------------
__global__ void reduce_nll(const int* __restrict__ target,
                           const float* __restrict__ part,
                           float* __restrict__ out) {
  const int i = blockIdx.x * blockDim.x + threadIdx.x;
  if (i >= NTOK) return;

  float m = -INFINITY, s = 0.f, tl = -INFINITY;
  float rc[3] = {-INFINITY, -INFINITY, -INFINITY};
  for (int c = 0; c < NCHUNK; ++c) {
    const float* p = part + ((size_t)c * NTOK + i) * 6;
    online_merge(m, s, p[0], p[1]);
    tl = fmaxf(tl, p[2]);
    rc[0] = fmaxf(rc[0], p[3]);
    rc[1] = fmaxf(rc[1], p[4]);
    rc[2] = fmaxf(rc[2], p[5]);
  }
  const float logZh = m + logf(s);

  const int t = target[i];
  float lp;
  if (t < 20000) {
    lp = tl - logZh;
  } else {
    int ci;
    if      (t <  40000) ci = 1;
    else if (t < 200000) ci = 2;
    else                 ci = 3;
    float mc = -INFINITY, sc = 0.f, tc = -INFINITY;
    for (int c = 0; c < NCHUNK; ++c) {
      const float* p = part + (((size_t)ci * NCHUNK + c) * NTOK + i) * 6;
      online_merge(mc, sc, p[0], p[1]);
      tc = fmaxf(tc, p[2]);
    }
    lp = (rc[3 - ci] - logZh) + (tc - (mc + logf(sc)));
  }
  out[i] = -lp;
}

// ---------------------------------------------------------------------------
extern "C" void kernel_launch(void* const* d_in, const int* in_sizes, int n_in,
                              void* d_out, int out_size, void* d_ws, size_t ws_size,
                              hipStream_t stream) {
  (void)in_sizes; (void)n_in; (void)out_size; (void)ws_size;

  const float* hidden = (const float*)d_in[0];
  const int*   target = (const int*)d_in[1];
  const float* Wp[4] = {(const float*)d_in[2], (const float*)d_in[5],
                        (const float*)d_in[8], (const float*)d_in[11]};
  const float* Bp[4] = {(const float*)d_in[3], (const float*)d_in[6],
                        (const float*)d_in[9], (const float*)d_in[12]};
  const float* Pp[4] = {(const float*)d_in[4], (const float*)d_in[7],
                        (const float*)d_in[10], (const float*)d_in[13]};

  const int dArr[4]    = {1024, 256, 64, 16};
  const int szArr[4]   = {20003, 20000, 160000, 67735};
  const int baseArr[4] = {0, 20000, 40000, 200000};

  float* ws = (float*)d_ws;
  size_t ofs[4];
  ofs[0] = 0;
  for (int c = 1; c < 4; ++c) ofs[c] = ofs[c - 1] + (size_t)NTOK * dArr[c - 1];
  float* partials = ws + (ofs[3] + (size_t)NTOK * dArr[3]);

  // Phase A: projection GEMMs (h @ p_c)
  for (int c = 0; c < 4; ++c) {
    dim3 grid(NTOK / 16, dArr[c] / 16);
    proj_gemm<<<grid, 32, 0, stream>>>(hidden, Pp[c], ws + ofs[c], dArr[c]);
  }

  // Phase B: fused logits + online softmax per cluster
  dim3 gridB(NCHUNK, NTOK / 128);
  for (int c = 0; c < 4; ++c) {
    const int tiles = (szArr[c] + 15) / 16;
    const int ct    = (tiles + NCHUNK - 1) / NCHUNK;
    const int ishd  = (c == 0) ? 1 : 0;
    switch (c) {
      case 0:
        cluster_pass<1024><<<gridB, 128, 0, stream>>>(
            Wp[c], Bp[c], ws + ofs[c], target, partials,
            szArr[c], baseArr[c], tiles, ct, ishd, c * NCHUNK);
        break;
      case 1:
        cluster_pass<256><<<gridB, 128, 0, stream>>>(
            Wp[c], Bp[c], ws + ofs[c], target, partials,
            szArr[c], baseArr[c], tiles, ct, ishd, c * NCHUNK);
        break;
      case 2:
        cluster_pass<64><<<gridB, 128, 0, stream>>>(
            Wp[c], Bp[c], ws + ofs[c], target, partials,
            szArr[c], baseArr[c], tiles, ct, ishd, c * NCHUNK);
        break;
      default:
        cluster_pass<16><<<gridB, 128, 0, stream>>>(
            Wp[c], Bp[c], ws + ofs[c], target, partials,
            szArr[c], baseArr[c], tiles, ct, ishd, c * NCHUNK);
        break;
    }
  }

  // Phase C: reduce -> NLL
  reduce_nll<<<dim3(4), dim3(256), 0, stream>>>(target, partials, (float*)d_out);
}